// JointLikelyhood_Gumbel_3504693313755
// MI455X (gfx1250) — compile-verified
//
#include <hip/hip_runtime.h>
#include <math.h>

// ---------------------------------------------------------------------------
// JointLikelyhood_Gumbel for MI455X (gfx1250, wave32).
//
// Stage 1: one wave32 per ROW (16384 waves ~ 16/SIMD for latency hiding).
//          NB CDF via log-domain pmf recurrence expressed as a lane prefix
//          scan: each chunk of 32 k-values -> per-lane term, 5-step shfl_up
//          inclusive scan, exp, accumulate. Wave-uniform loop bound (no lane
//          imbalance). Exact logpmf(y) via lgammaf (once per dim). Both dims
//          processed in-wave, copula combined locally; 8 rows/block reduced
//          in LDS -> 2048 block partials.
// Stage 2: single wave reduces the 2048 partials with V_WMMA_F32_16X16X4_F32
//          (A = ones => D rows are column sums; layout-agnostic, full fp32
//          precision), then a shfl fold.
// ---------------------------------------------------------------------------

typedef __attribute__((ext_vector_type(2))) float v2f;
typedef __attribute__((ext_vector_type(8))) float v8f;

#define NB_EPS 1e-6f

// Wave-cooperative: CDF(y) (sum of pmf over k<=y) and exact logpmf(y).
// All 32 lanes participate; y/rv/p1 are wave-uniform.
__device__ __forceinline__
void nb_cdf_logpmf_wave(float rv, float p1, int y, int lane,
                        float logp_p, float l1mp,
                        float& cdf_out, float& logp_out)
{
    // term(j) = log p + log((j-1+r)/j), j>=1 ; term(0) = 0
    // L(k) = r*log1p(-p) + sum_{j<=k} term(j)  (prefix sum -> lane scan)
    float carry = rv * l1mp;        // L(-1 of next chunk) seed = L0 basis
    float acc   = 0.0f;             // per-lane partial CDF

    for (int base = 0; base <= y; base += 32) {   // wave-uniform trip count
        const int   k  = base + lane;
        const float kf = (float)k;
        float t = (k == 0) ? 0.0f
                           : logp_p + logf((kf - 1.0f + rv) / kf);
        // inclusive prefix scan across the wave (wave32)
        #pragma unroll
        for (int off = 1; off < 32; off <<= 1) {
            const float n = __shfl_up(t, off);
            if (lane >= off) t += n;
        }
        const float L = carry + t;                // L(k) = logpmf(k)
        if (k <= y) acc += expf(L);
        carry = __shfl(L, 31);                    // L at end of chunk
    }

    // wave tree-reduce acc (all lanes end with the total)
    #pragma unroll
    for (int off = 16; off > 0; off >>= 1) acc += __shfl_xor(acc, off);
    cdf_out = acc;

    // exact log-pmf at y (3 lgammaf, once; uniform across lanes)
    const float yf = (float)y;
    logp_out = lgammaf(yf + rv) - lgammaf(yf + 1.0f) - lgammaf(rv)
             + rv * l1mp + yf * logp_p;
}

__global__ __launch_bounds__(256)
void nb_gumbel_stage1(const float* __restrict__ r,
                      const float* __restrict__ p,
                      const int*   __restrict__ target,
                      float*       __restrict__ partials)
{
    const int waveInBlk = threadIdx.x >> 5;               // 0..7
    const int lane      = threadIdx.x & 31;
    const int row       = blockIdx.x * 8 + waveInBlk;     // 0..16383

    // --- parameter transforms (wave-uniform scalars, broadcast loads) ---
    const float p1    = fminf(fmaxf(tanhf(p[2 * row]), 1e-4f), 0.9999f);
    const float theta = fmaxf(fmaxf(p[2 * row + 1], 0.0f) + 1.0f, 1.00001f);
    const float logp_p = logf(p1);
    const float l1mp   = log1pf(-p1);

    float t_sum = 0.0f, logp_sum = 0.0f;
    #pragma unroll
    for (int d = 0; d < 2; ++d) {
        const float rv = fmaxf(r[2 * row + d], 1e-4f);
        const int   y  = target[2 * row + d];
        float cdf, logp;
        nb_cdf_logpmf_wave(rv, p1, y, lane, logp_p, l1mp, cdf, logp);
        const float u = fminf(fmaxf(cdf, NB_EPS), 1.0f - NB_EPS);
        t_sum    += powf(-logf(u), theta);     // (-log u_d)^theta
        logp_sum += logp;
    }

    // Gumbel copula log-density term + row log-likelihood
    const float ll = logp_sum - powf(t_sum, 1.0f / theta);

    // --- block combine: 8 rows per block ---
    __shared__ float sll[8];
    if (lane == 0) sll[waveInBlk] = ll;
    __syncthreads();
    if (threadIdx.x == 0) {
        float s = 0.0f;
        #pragma unroll
        for (int i = 0; i < 8; ++i) s += sll[i];
        partials[blockIdx.x] = s;
    }
}

// Single-wave final reduction using fp32 WMMA (16x16x4, f32 in/out).
// A = all-ones (16x4)  =>  D[i][j] = sum_k B[k][j]; summing row 0's 16
// columns yields the total of all 64 B elements per chunk, independent of
// the per-lane matrix layout. C accumulates across chunks in fp32.
__global__ __launch_bounds__(32)
void nb_gumbel_stage2_wmma(const float* __restrict__ partials,
                           float*       __restrict__ out,
                           int nPartials, float invB)
{
    const int lane = threadIdx.x;   // 0..31, EXEC all ones (no divergence)

    v2f a; a.x = 1.0f; a.y = 1.0f;  // A = ones
    v8f c = {};                     // C = 0 accumulator

    for (int base = 0; base < nPartials; base += 64) {
        v2f b;
        b.x = partials[base + 2 * lane];
        b.y = partials[base + 2 * lane + 1];
        // 8 args: (neg_a, A, neg_b, B, c_mod, C, reuse_a, reuse_b)
        c = __builtin_amdgcn_wmma_f32_16x16x4_f32(
                false, a, false, b, (short)0, c, false, false);
    }

    // D row 0 lives in c[0] on lanes 0..15 (lanes 16..31 hold identical
    // row M=8). Fold the 16 column sums within each 16-lane group.
    float v = c[0];
    v += __shfl_xor(v, 1);
    v += __shfl_xor(v, 2);
    v += __shfl_xor(v, 4);
    v += __shfl_xor(v, 8);

    if (lane == 0) out[0] = -v * invB;   // -mean(ll)
}

extern "C" void kernel_launch(void* const* d_in, const int* in_sizes, int n_in,
                              void* d_out, int out_size, void* d_ws, size_t ws_size,
                              hipStream_t stream)
{
    (void)in_sizes; (void)n_in; (void)out_size; (void)ws_size;

    const float* r   = (const float*)d_in[0];   // (B,2) f32
    const float* p   = (const float*)d_in[1];   // (B,2) f32
    const int*   tgt = (const int*)  d_in[2];   // (B,2) int
    float* out = (float*)d_out;                 // 1 f32
    float* ws  = (float*)d_ws;                  // >= 2048 floats scratch

    constexpr int B       = 16384;
    constexpr int THREADS = 256;                // 8 wave32 per block
    constexpr int NBLOCKS = B / 8;              // one wave per row -> 2048

    nb_gumbel_stage1<<<NBLOCKS, THREADS, 0, stream>>>(r, p, tgt, ws);
    nb_gumbel_stage2_wmma<<<1, 32, 0, stream>>>(ws, out, NBLOCKS,
                                                1.0f / (float)B);
}